// DeepAggregateAutoEncoder_77781857731251
// MI455X (gfx1250) — compile-verified
//
#include <hip/hip_runtime.h>
#include <stdint.h>

// DeepAggregateAutoEncoder: 4 fused gather+min/max layers, batch-tiled in LDS.
// B=4096, layers 1024 -> 2048 -> 512 -> 2048 -> 1024, K=32 connections.

#define ROWS    8                    // batch rows per block (LDS-resident)
#define STRIDE  2049                 // odd row stride in floats -> conflict-free banks
#define THREADS 512                  // 16 waves (wave32)
#define NGROUP  (THREADS / ROWS)     // 64 feature groups

// ---- gfx1250 async global->LDS path (guarded; falls back to plain copy) ----
#if defined(__HIP_DEVICE_COMPILE__) && defined(__gfx1250__) && \
    __has_builtin(__builtin_amdgcn_global_load_async_to_lds_b128) && \
    __has_builtin(__builtin_amdgcn_s_wait_asynccnt)
#define USE_ASYNC_LDS 1
#else
#define USE_ASYNC_LDS 0
#endif

#if USE_ASYNC_LDS
// Builtin signature (from hipcc diagnostic + upstream def "vV4i*1V4i*3IiIi"):
//   void __builtin_amdgcn_global_load_async_to_lds_b128(
//       v4i addrspace(1)* src, v4i addrspace(3)* lds, imm int offset, imm int cpol)
typedef int v4i __attribute__((ext_vector_type(4)));
typedef __attribute__((address_space(1))) v4i* gv4i_ptr;   // pointer to AS1 v4i
typedef __attribute__((address_space(3))) v4i* lv4i_ptr;   // pointer to AS3 v4i

__device__ __forceinline__ gv4i_ptr to_global_v4(const void* p) {
    // generic global pointer value == full 64-bit VA; int->ptr cast crosses AS
    return (gv4i_ptr)(unsigned long long)(p);
}
__device__ __forceinline__ lv4i_ptr to_lds_v4(const void* p) {
    // low 32 bits of a generic shared-aperture address are the LDS byte offset
    return (lv4i_ptr)(unsigned)(unsigned long long)(p);
}
#endif

// One aggregation layer: gather K=32 features from LDS row, reduce min & max,
// select by op. lin/lout point into __shared__ arrays (AS3 recovered by inlining).
__device__ __forceinline__ void aggregate_layer(const float* __restrict__ lin,
                                                float* __restrict__ lout,
                                                const int* __restrict__ conn,
                                                const int* __restrict__ op,
                                                int nOut, int base, int group)
{
    for (int f = group; f < nOut; f += NGROUP) {
        const int4* c4 = (const int4*)(conn + (size_t)f * 32);
        // prefetch this group's next conn row into cache (speculative; OOB dropped)
        __builtin_prefetch(conn + (size_t)(f + NGROUP) * 32, 0, 1);

        float mn = __builtin_inff();
        float mx = -__builtin_inff();
#pragma unroll
        for (int k = 0; k < 8; ++k) {
            int4 c = c4[k];
            float a = lin[base + c.x];
            float b = lin[base + c.y];
            float d = lin[base + c.z];
            float e = lin[base + c.w];
            mn = fminf(mn, fminf(fminf(a, b), fminf(d, e)));
            mx = fmaxf(mx, fmaxf(fmaxf(a, b), fmaxf(d, e)));
        }
        lout[base + f] = (op[f] == 1) ? mx : mn;
    }
}

__global__ __launch_bounds__(THREADS)
void daae_fused_kernel(const float* __restrict__ x,
                       const int* __restrict__ conn0, const int* __restrict__ op0,
                       const int* __restrict__ conn1, const int* __restrict__ op1,
                       const int* __restrict__ conn2, const int* __restrict__ op2,
                       const int* __restrict__ conn3, const int* __restrict__ op3,
                       float* __restrict__ out)
{
    // Two ping-pong row buffers, odd stride for bank-conflict-free row access.
    __shared__ float bufA[ROWS * STRIDE];   // holds widths up to 2048
    __shared__ float bufB[ROWS * STRIDE];

    const int tid     = threadIdx.x;
    const int rowBase = blockIdx.x * ROWS;

    // ---- Stage input tile: ROWS x 1024 floats of x into bufA ----
    {
        const int lr = tid >> 6;     // 0..7  : row
        const int ln = tid & 63;     // 0..63 : 64 threads per row
        const float4* src = (const float4*)(x + (size_t)(rowBase + lr) * 1024);
        float* dst = bufA + lr * STRIDE;
#pragma unroll
        for (int j = 0; j < 256; j += 64) {   // 256 float4 per row / 64 threads
#if USE_ASYNC_LDS
            __builtin_amdgcn_global_load_async_to_lds_b128(
                to_global_v4(src + j + ln),
                to_lds_v4(dst + (size_t)(j + ln) * 4),
                0, 0);
#else
            float4 v = src[j + ln];
            *(float4*)(dst + (size_t)(j + ln) * 4) = v;
#endif
        }
    }
#if USE_ASYNC_LDS
    __builtin_amdgcn_s_wait_asynccnt(0);
#endif
    __syncthreads();

    const int row   = tid & (ROWS - 1);   // lane-major rows -> distinct LDS banks
    const int group = tid >> 3;           // 64 feature groups
    const int base  = row * STRIDE;

    // Layer pipeline, fully LDS-resident: A(1024) -> B(2048) -> A(512) -> B(2048) -> A(1024)
    aggregate_layer(bufA, bufB, conn0, op0, 2048, base, group);
    __syncthreads();
    aggregate_layer(bufB, bufA, conn1, op1, 512, base, group);
    __syncthreads();
    aggregate_layer(bufA, bufB, conn2, op2, 2048, base, group);
    __syncthreads();
    aggregate_layer(bufB, bufA, conn3, op3, 1024, base, group);
    __syncthreads();

    // ---- Coalesced write-out: ROWS x 1024 floats from bufA ----
    {
        const int lr = tid >> 6;
        const int ln = tid & 63;
        const float* srcl = bufA + lr * STRIDE;
        float4* dst = (float4*)(out + (size_t)(rowBase + lr) * 1024);
#pragma unroll
        for (int j = 0; j < 256; j += 64)
            dst[j + ln] = *(const float4*)(srcl + (size_t)(j + ln) * 4);
    }
}

extern "C" void kernel_launch(void* const* d_in, const int* in_sizes, int n_in,
                              void* d_out, int out_size, void* d_ws, size_t ws_size,
                              hipStream_t stream) {
    // setup_inputs() order: x, conn0, op0, conn1, op1, conn2, op2, conn3, op3
    const float* x     = (const float*)d_in[0];
    const int*   conn0 = (const int*)  d_in[1];
    const int*   op0   = (const int*)  d_in[2];
    const int*   conn1 = (const int*)  d_in[3];
    const int*   op1   = (const int*)  d_in[4];
    const int*   conn2 = (const int*)  d_in[5];
    const int*   op2   = (const int*)  d_in[6];
    const int*   conn3 = (const int*)  d_in[7];
    const int*   op3   = (const int*)  d_in[8];
    float*       out   = (float*)d_out;

    const int B = 4096;
    dim3 grid(B / ROWS);      // 512 blocks
    dim3 block(THREADS);      // 512 threads = 16 waves
    daae_fused_kernel<<<grid, block, 0, stream>>>(x, conn0, op0, conn1, op1,
                                                  conn2, op2, conn3, op3, out);
}